// GraphSAGE_31817117729422
// MI455X (gfx1250) — compile-verified
//
#include <hip/hip_runtime.h>

// GraphSAGE 2-layer inference for MI455X (gfx1250, wave32).
// Sparse aggregation = atomic-f32 scatter (memory bound, ~1.6GB/layer; the
// 51MB accumulator resolves in the 192MB L2).
// Dense GEMMs (~9.8 GFLOP total) = V_WMMA_F32_16X16X4_F32, fp32-exact.

typedef __attribute__((ext_vector_type(2))) float v2f;
typedef __attribute__((ext_vector_type(8))) float v8f;

#define N_NODES 100000
#define D_FEAT  128   // both D_IN and D_H are 128 (aggregation feature width)

// ---------------------------------------------------------------------------
// Zero a float buffer (n must be a multiple of 4). Grid-stride float4 stores.
// ---------------------------------------------------------------------------
__global__ void sage_zero_f32(float* __restrict__ p, long n) {
    long i      = ((long)blockIdx.x * blockDim.x + threadIdx.x) * 4;
    long stride = (long)gridDim.x * blockDim.x * 4;
    float4 z = {0.0f, 0.0f, 0.0f, 0.0f};
    for (long j = i; j < n; j += stride) {
        *(float4*)(p + j) = z;
    }
}

// ---------------------------------------------------------------------------
// Edge scatter: one wave per edge; each lane moves 4 consecutive floats of the
// 128-wide feature row (32 lanes x float4 = 128 floats), atomically added into
// msg[dst]. Lane 0 optionally counts in-degree.
// ---------------------------------------------------------------------------
template <bool ADD_DEG>
__global__ void sage_scatter(const int* __restrict__ src,
                             const int* __restrict__ dst,
                             const float* __restrict__ feat,
                             float* __restrict__ msg,
                             float* __restrict__ deg,
                             int E) {
    int wave   = ((int)blockIdx.x * blockDim.x + threadIdx.x) >> 5;
    int lane   = threadIdx.x & 31;
    int nwaves = ((int)gridDim.x * blockDim.x) >> 5;
    for (int e = wave; e < E; e += nwaves) {
        int s = src[e];
        int d = dst[e];
        const float4 v = *(const float4*)(feat + (size_t)s * D_FEAT + lane * 4);
        float* mrow = msg + (size_t)d * D_FEAT + lane * 4;
        atomicAdd(mrow + 0, v.x);
        atomicAdd(mrow + 1, v.y);
        atomicAdd(mrow + 2, v.z);
        atomicAdd(mrow + 3, v.w);
        if (ADD_DEG && lane == 0) atomicAdd(deg + d, 1.0f);
    }
}

// ---------------------------------------------------------------------------
// Fused SAGEConv GEMM:
//   out[m, :] = act( (msg[m,:]/max(deg[m],1)) @ Wl + bias + xroot[m,:] @ Wr )
//
// One 16x16 output tile per wave via V_WMMA_F32_16X16X4_F32.
// Block = (NOUT/16) waves covering all output columns of one 16-row M tile.
// K fully unrolled (K/4 * 2 = 64 WMMAs); 1/deg mean-scale fused into the
// A-fragment load (v_pk_mul_f32 feeding the WMMA).
//
// Fragment layouts (ISA 7.12.2, wave32):
//   A (16x4 f32):  lanes 0-15 hold M=lane, K = k0 / k0+1 in v0/v1;
//                  lanes 16-31 hold M=lane-16, K = k0+2 / k0+3.
//   B (4x16 f32):  symmetric, indexed by column N = lane&15.
//   C/D (16x16):   VGPR i -> row (i + 8*(lane>>4)), col (lane&15).
// 100000 % 16 == 0 so all tiles are full: EXEC stays all-1s (WMMA requires it).
// ---------------------------------------------------------------------------
template <int K, int NOUT, bool RELU>
__global__ void sage_fused_gemm(const float* __restrict__ msg,
                                const float* __restrict__ deg,
                                const float* __restrict__ xroot,
                                const float* __restrict__ Wl,
                                const float* __restrict__ bias,
                                const float* __restrict__ Wr,
                                float* __restrict__ out) {
    const int mt   = blockIdx.x;          // M tile (16 rows)
    const int nt   = threadIdx.x >> 5;    // wave id == N tile (16 cols)
    const int lane = threadIdx.x & 31;
    const int l16  = lane & 15;
    const int hi   = lane >> 4;

    const int arow = mt * 16 + l16;       // feature row this lane loads
    const float sc = 1.0f / fmaxf(deg[arow], 1.0f);
    const float* __restrict__ mrow = msg   + (size_t)arow * K;
    const float* __restrict__ xrow = xroot + (size_t)arow * K;
    const int bcol = nt * 16 + l16;       // weight column this lane loads

    v8f acc = {};
#pragma unroll
    for (int k0 = 0; k0 < K; k0 += 4) {
        const int ka = k0 + 2 * hi;
        v2f a, b;
        // aggregated-mean path: A = msg/deg, B = Wl
        a[0] = mrow[ka] * sc;
        a[1] = mrow[ka + 1] * sc;
        b[0] = Wl[ka * NOUT + bcol];
        b[1] = Wl[(ka + 1) * NOUT + bcol];
        acc = __builtin_amdgcn_wmma_f32_16x16x4_f32(false, a, false, b,
                                                    (short)0, acc, false, false);
        // root path: A = x, B = Wr
        a[0] = xrow[ka];
        a[1] = xrow[ka + 1];
        b[0] = Wr[ka * NOUT + bcol];
        b[1] = Wr[(ka + 1) * NOUT + bcol];
        acc = __builtin_amdgcn_wmma_f32_16x16x4_f32(false, a, false, b,
                                                    (short)0, acc, false, false);
    }

    const float bv = bias[bcol];
#pragma unroll
    for (int i = 0; i < 8; ++i) {
        const int orow = mt * 16 + i + 8 * hi;
        float v = acc[i] + bv;
        if (RELU) v = fmaxf(v, 0.0f);
        out[(size_t)orow * NOUT + bcol] = v;
    }
}

// ---------------------------------------------------------------------------
// inputs: x, W1l, b1, W1r, W2l, b2, W2r, edge_index   (setup_inputs order)
// workspace: msg[N*128] | deg[N] | h[N*128]  (~103 MB)
// ---------------------------------------------------------------------------
extern "C" void kernel_launch(void* const* d_in, const int* in_sizes, int n_in,
                              void* d_out, int out_size, void* d_ws, size_t ws_size,
                              hipStream_t stream) {
    const float* x   = (const float*)d_in[0];
    const float* W1l = (const float*)d_in[1];
    const float* b1  = (const float*)d_in[2];
    const float* W1r = (const float*)d_in[3];
    const float* W2l = (const float*)d_in[4];
    const float* b2  = (const float*)d_in[5];
    const float* W2r = (const float*)d_in[6];
    const int*  eidx = (const int*)d_in[7];
    const int E = in_sizes[7] / 2;          // edge_index is (2, E) row-major
    const int* src = eidx;
    const int* dst = eidx + E;

    float* ws  = (float*)d_ws;
    float* msg = ws;                                    // N*128 floats
    float* deg = msg + (size_t)N_NODES * D_FEAT;        // N floats
    float* h   = deg + N_NODES;                         // N*128 floats

    const long n_msg_deg = (long)N_NODES * D_FEAT + N_NODES;  // 12,900,000 (%4==0)

    // Layer 1: mean-aggregate + fused GEMM + ReLU  -> h
    sage_zero_f32<<<2048, 256, 0, stream>>>(msg, n_msg_deg);
    sage_scatter<true><<<4096, 256, 0, stream>>>(src, dst, x, msg, deg, E);
    sage_fused_gemm<128, 128, true><<<N_NODES / 16, 8 * 32, 0, stream>>>(
        msg, deg, x, W1l, b1, W1r, h);

    // Layer 2: re-aggregate h (degrees unchanged) + fused GEMM -> d_out
    sage_zero_f32<<<2048, 256, 0, stream>>>(msg, (long)N_NODES * D_FEAT);
    sage_scatter<false><<<4096, 256, 0, stream>>>(src, dst, h, msg, deg, E);
    sage_fused_gemm<128, 64, false><<<N_NODES / 16, 4 * 32, 0, stream>>>(
        msg, deg, h, W2l, b2, W2r, (float*)d_out);
}